// GCN_dev_5446018532029
// MI455X (gfx1250) — compile-verified
//
#include <hip/hip_runtime.h>
#include <hip/hip_bf16.h>

typedef float v2f __attribute__((ext_vector_type(2)));
typedef float v8f __attribute__((ext_vector_type(8)));

#define IN_F 128
#define HID  64

// ---------------------------------------------------------------------------
// Zero the accumulator regions of the workspace
// ---------------------------------------------------------------------------
__global__ void zero_kernel(float* __restrict__ p, long long count) {
    long long i = (long long)blockIdx.x * blockDim.x + threadIdx.x;
    long long stride = (long long)gridDim.x * blockDim.x;
    for (; i < count; i += stride) p[i] = 0.0f;
}

// ---------------------------------------------------------------------------
// Degree computation: atomic +1 per edge endpoint
// ---------------------------------------------------------------------------
__global__ void degree_kernel(const int* __restrict__ src, const int* __restrict__ dst,
                              float* __restrict__ deg_out, float* __restrict__ deg_in, int E) {
    int e = blockIdx.x * blockDim.x + threadIdx.x;
    if (e >= E) return;
    atomicAdd(&deg_out[src[e]], 1.0f);
    atomicAdd(&deg_in[dst[e]], 1.0f);
}

// In-place: deg -> clip(deg,1)^-0.5
__global__ void norm_kernel(float* __restrict__ a, float* __restrict__ b, int n) {
    int i = blockIdx.x * blockDim.x + threadIdx.x;
    if (i >= n) return;
    a[i] = rsqrtf(fmaxf(a[i], 1.0f));
    b[i] = rsqrtf(fmaxf(b[i], 1.0f));
}

// ---------------------------------------------------------------------------
// t1 = (x @ W1) * norm_src[:,None]    [N,64], fp32 WMMA 16x16x4
// Block = 128 threads = 4 waves; wave w owns the 16x16 tile at cols [16w,16w+16)
// for rows [16*blockIdx.x, +16). K loop: 128 / 4 = 32 WMMA ops.
// ---------------------------------------------------------------------------
__global__ __launch_bounds__(128) void gemm1_wmma_kernel(
    const float* __restrict__ x,        // [N,128]
    const float* __restrict__ W1,       // [128,64]
    const float* __restrict__ norm_src, // [N]
    float* __restrict__ t1,             // [N,64]
    int n)
{
    const int wave    = threadIdx.x >> 5;          // 0..3 -> column tile
    const int lane    = threadIdx.x & 31;
    const int halfsel = lane >> 4;                 // 0: K pair {0,1}, 1: K pair {2,3}
    const int l16     = lane & 15;
    const int rowbase = blockIdx.x * 16;
    if (rowbase >= n) return;
    const int col0 = wave * 16;

    // A loads: row = rowbase + l16 (same for both lane halves), clamp for safety
    const int arow = min(rowbase + l16, n - 1);
    const float* __restrict__ xrow = x + (size_t)arow * IN_F;

    v8f acc = {};
    #pragma unroll 4
    for (int k = 0; k < IN_F; k += 4) {
        const int ka = k + 2 * halfsel;            // this lane-half's K pair
        v2f a;
        a.x = xrow[ka + 0];
        a.y = xrow[ka + 1];
        v2f b;
        b.x = W1[(size_t)(ka + 0) * HID + col0 + l16];
        b.y = W1[(size_t)(ka + 1) * HID + col0 + l16];
        // 8 args: (neg_a, A, neg_b, B, c_mod, C, reuse_a, reuse_b)
        acc = __builtin_amdgcn_wmma_f32_16x16x4_f32(
            false, a, false, b, (short)0, acc, false, false);
    }

    // C/D layout: element acc[v] is C[M = v + 8*halfsel][N = l16]
    float* __restrict__ tbase = t1 + (size_t)(rowbase + 8 * halfsel) * HID + col0 + l16;
    const float* __restrict__ nbase = norm_src + rowbase + 8 * halfsel;

    if (rowbase + 16 <= n) {
        // Full tile: straight-line, no exec-mask churn.
        float nv[8];
        #pragma unroll
        for (int v = 0; v < 8; ++v) nv[v] = nbase[v];
        #pragma unroll
        for (int v = 0; v < 8; ++v) tbase[(size_t)v * HID] = acc[v] * nv[v];
    } else {
        // Ragged last tile (not hit when n % 16 == 0).
        #pragma unroll
        for (int v = 0; v < 8; ++v) {
            const int row = rowbase + v + 8 * halfsel;
            if (row < n) tbase[(size_t)v * HID] = acc[v] * nbase[v];
        }
    }
}

// ---------------------------------------------------------------------------
// agg1[dst[e], f] += t1[src[e], f]   — one lane per (edge, feature)
// ---------------------------------------------------------------------------
__global__ void scatter1_kernel(const int* __restrict__ src, const int* __restrict__ dst,
                                const float* __restrict__ t1, float* __restrict__ agg1,
                                long long total) {
    long long tid = (long long)blockIdx.x * blockDim.x + threadIdx.x;
    if (tid >= total) return;
    const int e = (int)(tid >> 6);        // /64
    const int f = (int)(tid & 63);
    const int s = src[e];
    const int d = dst[e];
    atomicAdd(&agg1[(size_t)d * HID + f], t1[(size_t)s * HID + f]);
}

// ---------------------------------------------------------------------------
// Fused layer-1 epilogue + layer-2 input projection (no h1 materialization):
//   s2[i] = norm_src[i] * sum_f relu(agg1[i,f]*norm_dst[i] + b1[f]) * mask[i,f] * W2[f]
// One wave per node; lane covers features f and f+32; in-wave shuffle reduce.
// ---------------------------------------------------------------------------
__global__ __launch_bounds__(256) void post_dot_kernel(
    const float* __restrict__ agg1,      // [N,64]
    const float* __restrict__ norm_src,  // [N]
    const float* __restrict__ norm_dst,  // [N]
    const float* __restrict__ b1,        // [64]
    const float* __restrict__ W2,        // [64,1]
    const float* __restrict__ drop_mask, // [N,64]
    float* __restrict__ s2,              // [N]
    int n)
{
    const int node = blockIdx.x * (blockDim.x >> 5) + (threadIdx.x >> 5);
    const int lane = threadIdx.x & 31;
    if (node >= n) return;

    const float nd = norm_dst[node];
    float acc = 0.0f;
    #pragma unroll
    for (int j = 0; j < 2; ++j) {
        const int f = lane + 32 * j;
        float h = agg1[(size_t)node * HID + f] * nd + b1[f];
        h = fmaxf(h, 0.0f) * drop_mask[(size_t)node * HID + f];
        acc += h * W2[f];
    }
    #pragma unroll
    for (int off = 16; off > 0; off >>= 1)
        acc += __shfl_down(acc, off, 32);
    if (lane == 0) s2[node] = acc * norm_src[node];
}

// ---------------------------------------------------------------------------
// agg2[dst[e]] += s2[src[e]]
// ---------------------------------------------------------------------------
__global__ void scatter2_kernel(const int* __restrict__ src, const int* __restrict__ dst,
                                const float* __restrict__ s2, float* __restrict__ agg2, int E) {
    int e = blockIdx.x * blockDim.x + threadIdx.x;
    if (e >= E) return;
    atomicAdd(&agg2[dst[e]], s2[src[e]]);
}

// ---------------------------------------------------------------------------
// out[i] = sigmoid(agg2[i]*norm_dst[i] + b2)
// ---------------------------------------------------------------------------
__global__ void final_kernel(const float* __restrict__ agg2, const float* __restrict__ norm_dst,
                             const float* __restrict__ b2, float* __restrict__ out, int n) {
    int i = blockIdx.x * blockDim.x + threadIdx.x;
    if (i >= n) return;
    const float z = agg2[i] * norm_dst[i] + b2[0];
    out[i] = 1.0f / (1.0f + expf(-z));
}

// ---------------------------------------------------------------------------
extern "C" void kernel_launch(void* const* d_in, const int* in_sizes, int n_in,
                              void* d_out, int out_size, void* d_ws, size_t ws_size,
                              hipStream_t stream) {
    const float* x         = (const float*)d_in[0];   // [N,128]
    const float* W1        = (const float*)d_in[1];   // [128,64]
    const float* b1        = (const float*)d_in[2];   // [64]
    const float* W2        = (const float*)d_in[3];   // [64,1]
    const float* b2        = (const float*)d_in[4];   // [1]
    const int*   src       = (const int*)d_in[5];     // [E]
    const int*   dst       = (const int*)d_in[6];     // [E]
    const float* drop_mask = (const float*)d_in[7];   // [N,64]

    const int N = in_sizes[0] / IN_F;
    const int E = in_sizes[5];

    // Workspace layout (floats):
    float* f        = (float*)d_ws;
    float* deg_out  = f;                         // N   -> becomes norm_src
    float* deg_in   = f + (size_t)N;             // N   -> becomes norm_dst
    float* t1       = f + (size_t)2 * N;         // N*64
    float* agg1     = t1 + (size_t)N * HID;      // N*64
    float* s2       = agg1 + (size_t)N * HID;    // N
    float* agg2     = s2 + (size_t)N;            // N

    float* out = (float*)d_out;

    // 1. zero accumulators: deg_out+deg_in (2N) and agg1+s2+agg2 (64N+2N)
    {
        long long c1 = (long long)2 * N;
        zero_kernel<<<2048, 256, 0, stream>>>(deg_out, c1);
        long long c2 = (long long)N * HID + (long long)2 * N;
        zero_kernel<<<4096, 256, 0, stream>>>(agg1, c2);
    }
    // 2. degrees
    degree_kernel<<<(E + 255) / 256, 256, 0, stream>>>(src, dst, deg_out, deg_in, E);
    // 3. norms (in place)
    norm_kernel<<<(N + 255) / 256, 256, 0, stream>>>(deg_out, deg_in, N);
    const float* norm_src = deg_out;
    const float* norm_dst = deg_in;
    // 4. t1 = (x @ W1) * norm_src  (WMMA f32, exact fp32 semantics)
    gemm1_wmma_kernel<<<(N + 15) / 16, 128, 0, stream>>>(x, W1, norm_src, t1, N);
    // 5. edge aggregation, 64-wide
    {
        long long total = (long long)E * HID;
        long long blocks = (total + 255) / 256;
        scatter1_kernel<<<(unsigned)blocks, 256, 0, stream>>>(src, dst, t1, agg1, total);
    }
    // 6. fused relu/dropout/(h1@W2)*norm_src
    post_dot_kernel<<<(N + 7) / 8, 256, 0, stream>>>(agg1, norm_src, norm_dst, b1, W2,
                                                     drop_mask, s2, N);
    // 7. scalar edge aggregation
    scatter2_kernel<<<(E + 255) / 256, 256, 0, stream>>>(src, dst, s2, agg2, E);
    // 8. sigmoid epilogue
    final_kernel<<<(N + 255) / 256, 256, 0, stream>>>(agg2, norm_dst, b2, out, N);
}